// ResGCN_41068477284994
// MI455X (gfx1250) — compile-verified
//
#include <hip/hip_runtime.h>
#include <hip/hip_bf16.h>

// ---------------------------------------------------------------------------
// ResGCN forward for MI455X (gfx1250, wave32).
// - GEMMs: V_WMMA_F32_16X16X4_F32 (full fp32; pipeline is HBM-bound, so
//   low-precision WMMA buys nothing — keep exact numerics).
// - Weight staging: TDM tensor_load_to_lds (async DMA, TENSORcnt) into 64KB
//   of the WGP's 320KB LDS, issued once per block by wave 0.
// - Edge aggregation: float4 gathers + fp32 global atomics (bandwidth limit).
// ---------------------------------------------------------------------------

typedef __attribute__((ext_vector_type(2))) float    v2f;
typedef __attribute__((ext_vector_type(8))) float    v8f;
typedef __attribute__((ext_vector_type(4))) unsigned u32x4;
typedef __attribute__((ext_vector_type(8))) unsigned u32x8;

#define HDIM 128
#define BN_FLAG   1
#define RELU_FLAG 2
#define BIAS_FLAG 4

// ---------------- small utility kernels ----------------

__global__ void fill_f(float* __restrict__ p, float v, int n) {
    int t = blockIdx.x * blockDim.x + threadIdx.x;
    if (t < n) p[t] = v;
}

__global__ void deg_edges(const int* __restrict__ row, float* __restrict__ deg, int E) {
    int e = blockIdx.x * blockDim.x + threadIdx.x;
    if (e < E) atomicAdd(&deg[row[e]], 1.0f);
}

__global__ void dinv_finish(float* __restrict__ deg, int n) {
    int t = blockIdx.x * blockDim.x + threadIdx.x;
    if (t < n) deg[t] = rsqrtf(deg[t]);
}

// per-column sum and sum-of-squares over M rows of a [M,128] matrix
__global__ void col_stats(const float* __restrict__ X, int M, float* __restrict__ stats) {
    int j = threadIdx.x;            // 128 threads
    float s = 0.f, s2 = 0.f;
    for (int r = blockIdx.x; r < M; r += gridDim.x) {
        float v = X[(size_t)r * HDIM + j];
        s += v; s2 += v * v;
    }
    atomicAdd(&stats[j], s);
    atomicAdd(&stats[HDIM + j], s2);
}

// fold BN (training mode, biased var) into y = x*scale + shift
__global__ void make_scaleshift(const float* __restrict__ stats, float invM,
                                const float* __restrict__ g, const float* __restrict__ b,
                                float* __restrict__ scale, float* __restrict__ shift) {
    int j = threadIdx.x;            // 128 threads, 1 block
    float m   = stats[j] * invM;
    float var = stats[HDIM + j] * invM - m * m;
    float sc  = g[j] * rsqrtf(var + 1e-5f);
    scale[j] = sc;
    shift[j] = b[j] - m * sc;
}

// out[i,:] = dinv[i]^2 * h2[i,:] + bias[:]   (float4 over 32 lanes per row)
__global__ void self_init(const float4* __restrict__ h2, const float* __restrict__ dinv,
                          const float* __restrict__ bias, float4* __restrict__ out, int nq) {
    int t = blockIdx.x * blockDim.x + threadIdx.x;
    if (t < nq) {
        int i = t >> 5, j4 = (t & 31) * 4;
        float d = dinv[i]; float d2 = d * d;
        float4 h = h2[t];
        float4 o;
        o.x = d2 * h.x + bias[j4 + 0];
        o.y = d2 * h.y + bias[j4 + 1];
        o.z = d2 * h.z + bias[j4 + 2];
        o.w = d2 * h.w + bias[j4 + 3];
        out[t] = o;
    }
}

// out[col[e],:] += dinv[row]*dinv[col] * h2[row[e],:]
// one edge per wave32; each lane handles 4 consecutive features (b128 gather)
__global__ void edge_scatter(const int* __restrict__ row, const int* __restrict__ col,
                             const float* __restrict__ dinv, const float* __restrict__ h2,
                             float* __restrict__ out, int E) {
    int e    = blockIdx.x * 8 + (threadIdx.x >> 5);
    int lane = threadIdx.x & 31;
    if (e < E) {
        int r = row[e], c = col[e];
        float ne = dinv[r] * dinv[c];
        float4 v = *(const float4*)(h2 + (size_t)r * HDIM + lane * 4);
        float* o = out + (size_t)c * HDIM + lane * 4;
        atomicAdd(o + 0, ne * v.x);
        atomicAdd(o + 1, ne * v.y);
        atomicAdd(o + 2, ne * v.z);
        atomicAdd(o + 3, ne * v.w);
    }
}

__global__ void relu4_k(float4* __restrict__ p, int nq) {
    int t = blockIdx.x * blockDim.x + threadIdx.x;
    if (t < nq) {
        float4 v = p[t];
        v.x = fmaxf(v.x, 0.f); v.y = fmaxf(v.y, 0.f);
        v.z = fmaxf(v.z, 0.f); v.w = fmaxf(v.w, 0.f);
        p[t] = v;
    }
}

__global__ void pool_k(const float* __restrict__ x, const int* __restrict__ batch,
                       float* __restrict__ xp, int n) {
    int t = blockIdx.x * blockDim.x + threadIdx.x;
    if (t < n) {
        int i = t >> 7, j = t & (HDIM - 1);
        atomicAdd(&xp[(size_t)batch[i] * HDIM + j], x[t]);
    }
}

__global__ void bn_apply(const float* __restrict__ x, const float* __restrict__ scale,
                         const float* __restrict__ shift, float* __restrict__ out, int n) {
    int t = blockIdx.x * blockDim.x + threadIdx.x;
    if (t < n) {
        int j = t & (HDIM - 1);
        out[t] = x[t] * scale[j] + shift[j];
    }
}

// ---------------- WMMA GEMM: Out[M,128] = op(A[M,128]) @ W[128,128] ----------------
// op(A) = A*scale + shift (per column) when BN_FLAG set (compile-time).
// One wave per 16x128 output strip (8 16x16 N-tiles); 8 M-tiles per 256-thread block.
// W (64KB contiguous) is DMA'd into LDS by the Tensor Data Mover, issued by
// wave 0; completion via TENSORcnt, then a workgroup barrier.
// Fragment layouts per CDNA5 ISA 7.12.2:
//   A (16x4 f32, MxK): lanes 0-15 -> M=lane, VGPR0/1 = K0/K1; lanes 16-31 -> K2/K3
//   B (4x16 f32, KxN): lanes 0-15 -> N=lane, VGPR0/1 = K0/K1; lanes 16-31 -> K2/K3
//   C/D (16x16 f32):  VGPR r: lanes 0-15 -> M=r, N=lane; lanes 16-31 -> M=r+8
template <int FLAGS>
__global__ __launch_bounds__(256)
void gemm128_wmma(const float* __restrict__ A, const float* __restrict__ W,
                  const float* __restrict__ scale, const float* __restrict__ shift,
                  const float* __restrict__ bias, float* __restrict__ Out, int M) {
    __shared__ float sW[HDIM * HDIM];           // 64 KB
    int tid = threadIdx.x;

    if (tid < 32) {                             // wave 0 drives the TDM
        unsigned           lds = (unsigned)(unsigned long long)(&sW[0]);
        unsigned long long ga  = (unsigned long long)W;
        // D# group 0: count=1 | lds_addr | global_addr[56:0] | type=2
        u32x4 g0 = { 1u, lds, (unsigned)ga, (unsigned)(ga >> 32) | (2u << 30) };
        // D# group 1: data_size=4B (dw0[17:16]=2); tensor_dim0=16384 elems
        // (dw1[31:16] low16, dw2[15:0] hi16=0); tensor_dim1=1 (dw2[31:16]);
        // tile_dim0=16384 (dw3[31:16]); tile_dim1=1 (dw4[15:0]);
        // tensor_dim0_stride=16384 (dw5).
        u32x8 g1 = { 0x00020000u, 0x40000000u, 0x00010000u, 0x40000000u,
                     0x00000001u, 0x00004000u, 0u, 0u };
        asm volatile("tensor_load_to_lds %0, %1" :: "s"(g0), "s"(g1) : "memory");
        __builtin_amdgcn_s_wait_tensorcnt(0);
    }
    __syncthreads();

    int wave = tid >> 5;
    int lane = tid & 31;
    int half = lane >> 4;                       // 0: K pair {0,1}, 1: K pair {2,3}
    int lr   = lane & 15;

    int mtile = blockIdx.x * 8 + wave;
    if (mtile * 16 >= M) return;

    const float* arow = A + (size_t)(mtile * 16 + lr) * HDIM;

    // preload all 32 A-fragments (K=0..127), fusing BN scale/shift
    v2f af[32];
#pragma unroll
    for (int kk = 0; kk < 32; ++kk) {
        int c = kk * 4 + half * 2;
        v2f a;
        a.x = arow[c];
        a.y = arow[c + 1];
        if (FLAGS & BN_FLAG) {
            a.x = a.x * scale[c]     + shift[c];
            a.y = a.y * scale[c + 1] + shift[c + 1];
        }
        af[kk] = a;
    }

    for (int n0 = 0; n0 < 8; ++n0) {            // 8 N-tiles of 16 columns
        v8f acc = {};
        int colb = n0 * 16 + lr;
#pragma unroll
        for (int kk = 0; kk < 32; ++kk) {
            int kr = kk * 4 + half * 2;
            v2f b;
            b.x = sW[kr * HDIM + colb];
            b.y = sW[(kr + 1) * HDIM + colb];
            acc = __builtin_amdgcn_wmma_f32_16x16x4_f32(
                false, af[kk], false, b, (short)0, acc, false, false);
        }
#pragma unroll
        for (int r = 0; r < 8; ++r) {
            int orow = mtile * 16 + half * 8 + r;
            int ocol = n0 * 16 + lr;
            float v = acc[r];
            if (FLAGS & BIAS_FLAG) v += bias[ocol];
            if (FLAGS & RELU_FLAG) v = fmaxf(v, 0.f);
            Out[(size_t)orow * HDIM + ocol] = v;
        }
    }
}

// ---------------- host-side orchestration ----------------

extern "C" void kernel_launch(void* const* d_in, const int* in_sizes, int n_in,
                              void* d_out, int out_size, void* d_ws, size_t ws_size,
                              hipStream_t stream) {
    const float* x          = (const float*)d_in[0];
    const int*   eidx       = (const int*)  d_in[1];   // [2,E]: rows then cols
    const int*   batch      = (const int*)  d_in[2];
    const float* gamma_feat = (const float*)d_in[3];
    const float* beta_feat  = (const float*)d_in[4];
    const float* W_feat     = (const float*)d_in[5];
    const float* bn_g       = (const float*)d_in[6];   // [L,128]
    const float* bn_b       = (const float*)d_in[7];
    const float* Wc         = (const float*)d_in[8];   // [L,128,128]
    const float* bc         = (const float*)d_in[9];   // [L,128]
    const float* bn_fc_g    = (const float*)d_in[10];
    const float* bn_fc_b    = (const float*)d_in[11];
    const float* W_fc       = (const float*)d_in[12];
    const float* b_fc       = (const float*)d_in[13];
    const float* bn_hg      = (const float*)d_in[14];
    const float* bn_hb      = (const float*)d_in[15];

    const int N = in_sizes[2];          // nodes (multiple of 16: 50000)
    const int E = in_sizes[1] / 2;      // edges
    const int G = out_size / HDIM;      // graphs (256)
    const int NF = N * HDIM;
    const int NQ = NF / 4;              // float4 elements
    const int L = 3;

    const int* erow = eidx;
    const int* ecol = eidx + E;

    // workspace carve-up (floats)
    float* ws    = (float*)d_ws;
    float* dinv  = ws;                                   size_t o = ((size_t)N + 63) & ~63ull;
    float* hA    = ws + o;                               o += (size_t)NF;
    float* hB    = ws + o;                               o += (size_t)NF;
    float* stats = ws + o;                               o += 256;
    float* scale = ws + o;                               o += HDIM;
    float* shift = ws + o;                               o += HDIM;
    float* xp    = ws + o;                               o += (size_t)G * HDIM;
    float* xp2   = ws + o;                               o += (size_t)G * HDIM;
    (void)ws_size; (void)n_in;

    const int T = 256;
    auto blocks = [](int n, int t) { return (n + t - 1) / t; };

    // 1) degrees (self loop = 1) -> dinv = deg^-0.5
    fill_f<<<blocks(N, T), T, 0, stream>>>(dinv, 1.0f, N);
    deg_edges<<<blocks(E, T), T, 0, stream>>>(erow, dinv, E);
    dinv_finish<<<blocks(N, T), T, 0, stream>>>(dinv, N);

    const int gemmGridN = (N / 16 + 7) / 8;   // 8 M-tiles per block
    const int gemmGridG = (G / 16 + 7) / 8;

    // 2) feat BN + conv_feat: hA = relu(BN(x) @ W_feat)
    fill_f<<<1, T, 0, stream>>>(stats, 0.f, 256);
    col_stats<<<256, HDIM, 0, stream>>>(x, N, stats);
    make_scaleshift<<<1, HDIM, 0, stream>>>(stats, 1.0f / N, gamma_feat, beta_feat, scale, shift);
    gemm128_wmma<BN_FLAG | RELU_FLAG><<<gemmGridN, T, 0, stream>>>(
        x, W_feat, scale, shift, nullptr, hA, N);

    // 3) GCN layers: hA = relu( scatter(norm * (BN(hA)@Wc)) + self_n*(BN(hA)@Wc) + bc )
    for (int l = 0; l < L; ++l) {
        fill_f<<<1, T, 0, stream>>>(stats, 0.f, 256);
        col_stats<<<256, HDIM, 0, stream>>>(hA, N, stats);
        make_scaleshift<<<1, HDIM, 0, stream>>>(stats, 1.0f / N,
                                                bn_g + l * HDIM, bn_b + l * HDIM, scale, shift);
        // hB = BN(hA) @ Wc[l]
        gemm128_wmma<BN_FLAG><<<gemmGridN, T, 0, stream>>>(
            hA, Wc + (size_t)l * HDIM * HDIM, scale, shift, nullptr, hB, N);
        // hA = self_n * hB + bc[l]     (hA's old contents fully consumed by GEMM)
        self_init<<<blocks(NQ, T), T, 0, stream>>>((const float4*)hB, dinv,
                                                   bc + l * HDIM, (float4*)hA, NQ);
        // hA[col] += norm_e * hB[row]
        edge_scatter<<<(E + 7) / 8, T, 0, stream>>>(erow, ecol, dinv, hB, hA, E);
        relu4_k<<<blocks(NQ, T), T, 0, stream>>>((float4*)hA, NQ);
    }

    // 4) global_add_pool -> xp [G,128]
    fill_f<<<blocks(G * HDIM, T), T, 0, stream>>>(xp, 0.f, G * HDIM);
    pool_k<<<blocks(NF, T), T, 0, stream>>>(hA, batch, xp, NF);

    // 5) FC: xp2 = relu(BN(xp) @ W_fc + b_fc)
    fill_f<<<1, T, 0, stream>>>(stats, 0.f, 256);
    col_stats<<<256, HDIM, 0, stream>>>(xp, G, stats);
    make_scaleshift<<<1, HDIM, 0, stream>>>(stats, 1.0f / G, bn_fc_g, bn_fc_b, scale, shift);
    gemm128_wmma<BN_FLAG | RELU_FLAG | BIAS_FLAG><<<gemmGridG, T, 0, stream>>>(
        xp, W_fc, scale, shift, b_fc, xp2, G);

    // 6) final BN -> d_out
    fill_f<<<1, T, 0, stream>>>(stats, 0.f, 256);
    col_stats<<<256, HDIM, 0, stream>>>(xp2, G, stats);
    make_scaleshift<<<1, HDIM, 0, stream>>>(stats, 1.0f / G, bn_hg, bn_hb, scale, shift);
    bn_apply<<<blocks(G * HDIM, T), T, 0, stream>>>(xp2, scale, shift, (float*)d_out, G * HDIM);
}